// TRIP_49452253447057
// MI455X (gfx1250) — compile-verified
//
#include <hip/hip_runtime.h>

// DRAW-style glimpse: out[b,c,12,12] = 4.0 * Fy(12x128) @ img[b,c](128x128) @ Fx^T(128x12)
// Memory-bound: 536 MB img stream @ 23.3 TB/s ~= 23 us floor. fp32 end-to-end; the small
// GEMMs use exact-fp32 V_WMMA_F32_16X16X4_F32 chains (wave32, 16x16 tiles, 12 padded to 16).
// img tiles are staged global->LDS with CDNA5 GLOBAL_LOAD_ASYNC_TO_LDS_B128 (ASYNCcnt path):
// no VGPR round-trip, coalesced lines; one slot per wave, each tile's fetch is waited on before its WMMA chain.

typedef float v2f __attribute__((ext_vector_type(2)));
typedef float v8f __attribute__((ext_vector_type(8)));
typedef int   v4i __attribute__((ext_vector_type(4)));

#define IMG_A 128
#define IMG_B 128
#define NN    12
#define PITCH 132                 // padded row stride (floats): conflict-free 16-lane column reads
#define TILE_FLOATS (16 * PITCH)  // 2112 floats = one padded 16x128 fp32 tile

typedef const __attribute__((address_space(1))) void* gas_cptr;
typedef __attribute__((address_space(3))) void*       las_ptr;
typedef __attribute__((address_space(1))) v4i*        gv4i_p;
typedef __attribute__((address_space(3))) v4i*        lv4i_p;

__device__ __forceinline__ void async_copy_b128(const float* g, float* l) {
#if __has_builtin(__builtin_amdgcn_global_load_async_to_lds_b128)
    gas_cptr gvoid = (gas_cptr)(const void*)g;   // generic -> AS1 addrspacecast
    las_ptr  lvoid = (las_ptr)(void*)l;          // generic -> AS3 addrspacecast
    __builtin_amdgcn_global_load_async_to_lds_b128((gv4i_p)gvoid, (lv4i_p)lvoid, 0, 0);
#else
    const unsigned lds_off = (unsigned)(size_t)(las_ptr)(void*)l;  // 32-bit LDS byte offset
    const unsigned long long gaddr = (unsigned long long)(size_t)g;
    asm volatile("global_load_async_to_lds_b128 %0, %1, off"
                 :: "v"(lds_off), "v"(gaddr) : "memory");
#endif
}

__device__ __forceinline__ void wait_async_zero() {
#if __has_builtin(__builtin_amdgcn_s_wait_asynccnt)
    __builtin_amdgcn_s_wait_asynccnt(0);
    asm volatile("" ::: "memory");
#else
    asm volatile("s_wait_asynccnt 0x0" ::: "memory");
#endif
}

// DS-pipe reads (DScnt) and async-LDS writes (ASYNCcnt) are unordered across pipelines.
// Fence before re-filling a slot whose previous contents may still have DS reads in flight.
__device__ __forceinline__ void wait_ds_zero() {
    asm volatile("s_wait_dscnt 0x0" ::: "memory");
}

__global__ __launch_bounds__(256) void draw_glimpse_kernel(
    const float* __restrict__ vec,   // (BATCH, 3)
    const float* __restrict__ img,   // (BATCH, 2, 128, 128)
    float*       __restrict__ out)   // (BATCH, 2, 12, 12)
{
    // 12 padded tiles: Fx, Fy, tmpT[2 channels], one img slot per wave (8 waves). ~99 KB:
    // two workgroups still co-resident per 320 KB WGP.
    __shared__ float ldsFx  [TILE_FLOATS];
    __shared__ float ldsFy  [TILE_FLOATS];
    __shared__ float ldsTmpT[2 * TILE_FLOATS];
    __shared__ float ldsImg [8 * TILE_FLOATS];

    const int b    = blockIdx.x;
    const int t    = threadIdx.x;
    const int wave = t >> 5;      // 0..7
    const int lane = t & 31;
    const int lrow = lane & 15;   // row index within 16 (M for A, N for B, C/D column)
    const int lhi  = lane >> 4;   // half-wave: K-offset +2 (A/B frags), M-offset +8 (C/D)

    // ---------------- Filterbank construction (per batch) ----------------
    const float gx_ = vec[b * 3 + 0];
    const float gy_ = vec[b * 3 + 1];
    const float ld  = vec[b * 3 + 2];
    const float delta = 4.0f * (ld + 1.0f);
    const float gx = (IMG_A * 0.5f) * (gx_ + 1.0f);
    const float gy = (IMG_B * 0.5f) * (gy_ + 1.0f);
    // sigma2 = 2 -> 1/(2*sigma2) = 0.25 ; gamma = 4

    for (int idx = t; idx < TILE_FLOATS; idx += 256) {
        const int n = idx / PITCH;
        const int w = idx - n * PITCH;
        float fx = 0.0f, fy = 0.0f;
        if (n < NN && w < IMG_B) {
            const float shift = ((float)n - (NN * 0.5f) - 0.5f) * delta;  // (n - 6.5)*delta
            const float dx = (float)w - (gx + shift);
            const float dy = (float)w - (gy + shift);
            fx = __expf(-dx * dx * 0.25f);
            fy = __expf(-dy * dy * 0.25f);
        }
        ldsFx[idx] = fx;   // rows 12..15 and cols 128..131 zeroed -> exact 16x16 padding
        ldsFy[idx] = fy;
    }
    __syncthreads();

    // Row-normalize: 16 threads per row, shuffle-reduce within 16-lane halves (wave32).
    {
        const int row = t >> 4;   // 0..15
        const int j   = t & 15;
        float sx = 0.0f, sy = 0.0f;
        for (int w = j; w < IMG_B; w += 16) {
            sx += ldsFx[row * PITCH + w];
            sy += ldsFy[row * PITCH + w];
        }
        #pragma unroll
        for (int m = 8; m; m >>= 1) {
            sx += __shfl_xor(sx, m, 32);
            sy += __shfl_xor(sy, m, 32);
        }
        const float ix = 1.0f / fmaxf(sx, 1e-8f);
        const float iy = 1.0f / fmaxf(sy, 1e-8f);
        for (int w = j; w < IMG_B; w += 16) {
            ldsFx[row * PITCH + w] *= ix;
            ldsFy[row * PITCH + w] *= iy;
        }
    }
    __syncthreads();

    // ---------------- Step 1: tmp[h][n] = sum_w img[h][w] * Fx[n][w] ----------------
    // Per wave: one 16-row h-tile per channel. A = img tile (16xK), B = Fx (B[w][n]=Fx[n][w]).
    // Both fragments read LDS at (row = lane%16, col = 4k + 2*(lane>>4)) -> one ds_load_b64 each.
    float*       imgSlot = ldsImg + wave * TILE_FLOATS;
    const float* aBase   = imgSlot + lrow * PITCH + lhi * 2;
    const float* bFx     = ldsFx   + lrow * PITCH + lhi * 2;

    for (int c = 0; c < 2; ++c) {
        const float* gsrc = img + (((size_t)b * 2 + c) * IMG_A + (size_t)wave * 16) * IMG_B;

        // Make sure no DS reads of this slot's previous tile are still in flight before
        // the async engine may overwrite it (DS vs ASYNC pipes are unordered).
        wait_ds_zero();

        // Async global -> LDS staging: 16 rows x (32 lanes * 16B) = 8 KB tile, no VGPR data.
        #pragma unroll
        for (int r = 0; r < 16; ++r) {
            async_copy_b128(gsrc + r * IMG_B + lane * 4,
                            imgSlot + r * PITCH + lane * 4);
        }
        wait_async_zero();   // s_wait_asynccnt 0: tile landed in LDS

        v8f acc = {};
        #pragma unroll
        for (int k = 0; k < 32; ++k) {
            const v2f a  = *(const v2f*)(aBase + 4 * k);
            const v2f bf = *(const v2f*)(bFx   + 4 * k);
            acc = __builtin_amdgcn_wmma_f32_16x16x4_f32(
                false, a, false, bf, (short)0, acc, false, false);
        }

        // D layout: VGPR r -> (M=r+8*lhi, N=lrow). Store transposed: tmpT[n][h].
        float* tT = ldsTmpT + c * TILE_FLOATS + lrow * PITCH + wave * 16 + lhi * 8;
        #pragma unroll
        for (int r = 0; r < 8; ++r) tT[r] = acc[r];
    }
    __syncthreads();

    // ---------------- Step 2: glimpse[m][n] = sum_h Fy[m][h] * tmp[h][n] ----------------
    if (wave < 2) {                       // wave-uniform branch: EXEC stays all-ones for WMMA
        const int c = wave;
        const float* aFy = ldsFy   + lrow * PITCH + lhi * 2;
        const float* bT  = ldsTmpT + c * TILE_FLOATS + lrow * PITCH + lhi * 2;

        v8f acc = {};
        #pragma unroll
        for (int k = 0; k < 32; ++k) {
            const v2f a  = *(const v2f*)(aFy + 4 * k);
            const v2f bf = *(const v2f*)(bT  + 4 * k);
            acc = __builtin_amdgcn_wmma_f32_16x16x4_f32(
                false, a, false, bf, (short)0, acc, false, false);
        }

        const int n = lrow;
        if (n < NN) {
            float* o = out + ((size_t)b * 2 + c) * (NN * NN);
            #pragma unroll
            for (int r = 0; r < 8; ++r) {
                const int m = r + lhi * 8;
                if (m < NN) o[m * NN + n] = 4.0f * acc[r];   // gamma = 4
            }
        }
    }
}

extern "C" void kernel_launch(void* const* d_in, const int* in_sizes, int n_in,
                              void* d_out, int out_size, void* d_ws, size_t ws_size,
                              hipStream_t stream) {
    const float* vec = (const float*)d_in[0];   // (BATCH, 3) fp32
    const float* img = (const float*)d_in[1];   // (BATCH, 2, 128, 128) fp32
    float* out = (float*)d_out;                 // (BATCH, 2, 12, 12) fp32

    const int batch = in_sizes[0] / 3;          // 4096
    draw_glimpse_kernel<<<dim3(batch), dim3(256), 0, stream>>>(vec, img, out);
}